// AGCN_Block_19739669692561
// MI455X (gfx1250) — compile-verified
//
#include <hip/hip_runtime.h>
#include <math.h>

// MI455X (gfx1250) implementation. fp32 WMMA (V_WMMA_F32_16X16X4_F32) for the two
// heavy GEMMs; everything fused to keep HBM traffic ~1GB (memory-bound block).
typedef __attribute__((ext_vector_type(2))) float v2f;
typedef __attribute__((ext_vector_type(8))) float v8f;

#define NB   64
#define CC   64
#define TDIM 300
#define VV   25
#define SS   3
#define TVS  (TDIM*VV)     // 7500
#define PT   16            // time steps per tile
#define NTIL 19            // ceil(300/16)
#define PPOS (PT*VV)       // 400 positions per tile
#define KGC  192           // S*C
#define KTC  576           // 9*C
#define NSLAB 16           // batch slabs for BN stats (4 batches each)

// ---------------------------------------------------------------- stage 1:
// xm = mean_T(x); theta/phi = W @ xm + b  (per-batch block)
__global__ void k_stage1(const float* __restrict__ x,
                         const float* __restrict__ wth, const float* __restrict__ bth,
                         const float* __restrict__ wph, const float* __restrict__ bph,
                         float* __restrict__ theta, float* __restrict__ phi)
{
    __shared__ float xm[CC*VV]; // 1600 floats
    const int n = blockIdx.x, tid = threadIdx.x;
    for (int i = tid; i < CC*VV; i += 256) {
        int c = i / VV, v = i % VV;
        const float* xp = x + (((size_t)n*CC + c)*TDIM)*VV + v;
        float s = 0.f;
        for (int t = 0; t < TDIM; ++t) s += xp[t*VV];
        xm[i] = s * (1.0f/TDIM);
    }
    __syncthreads();
    for (int i = tid; i < 2*CC*VV; i += 256) {
        int which = i / (CC*VV);
        int r = i % (CC*VV);
        int e = r / VV, v = r % VV;
        const float* w = which ? wph : wth;
        float s = which ? bph[e] : bth[e];
        #pragma unroll 8
        for (int c = 0; c < CC; ++c) s += w[e*CC + c] * xm[c*VV + v];
        (which ? phi : theta)[(size_t)n*CC*VV + r] = s;
    }
}

// ---------------------------------------------------------------- stage 2:
// sim = theta^T phi, softmax rows, adapt = A + B + C (per-batch block)
__global__ void k_adaptive(const float* __restrict__ theta, const float* __restrict__ phi,
                           const float* __restrict__ A, const float* __restrict__ Bm,
                           float* __restrict__ adapt)
{
    __shared__ float th[CC*VV], ph[CC*VV], sim[VV*VV];
    const int n = blockIdx.x, tid = threadIdx.x;
    for (int i = tid; i < CC*VV; i += 256) {
        th[i] = theta[(size_t)n*CC*VV + i];
        ph[i] = phi  [(size_t)n*CC*VV + i];
    }
    __syncthreads();
    for (int i = tid; i < VV*VV; i += 256) {
        int v = i / VV, w = i % VV;
        float s = 0.f;
        #pragma unroll 8
        for (int e = 0; e < CC; ++e) s += th[e*VV + v] * ph[e*VV + w];
        sim[i] = s;
    }
    __syncthreads();
    if (tid < VV) {
        float mx = -3.4e38f;
        for (int w = 0; w < VV; ++w) mx = fmaxf(mx, sim[tid*VV + w]);
        float ex[VV]; float sum = 0.f;
        for (int w = 0; w < VV; ++w) { ex[w] = expf(sim[tid*VV + w] - mx); sum += ex[w]; }
        float inv = 1.0f / sum;
        for (int w = 0; w < VV; ++w) sim[tid*VV + w] = ex[w] * inv;
    }
    __syncthreads();
    for (int i = tid; i < SS*VV*VV; i += 256) {
        int r = i % (VV*VV);
        adapt[(size_t)n*SS*VV*VV + i] = A[i] + Bm[i] + sim[r];
    }
}

// ---------------------------------------------------------------- prep:
// w_comb[o][k] = sum_s w_W[s*64+o][k]   (S-sum commutes with the linear map)
// wre[o][kt*64+c] = w_t[o][c][kt]        (implicit-GEMM friendly layout)
__global__ void k_prep(const float* __restrict__ wW, const float* __restrict__ wt,
                       float* __restrict__ wcomb, float* __restrict__ wre)
{
    int i = blockIdx.x*256 + threadIdx.x;
    if (i < CC*KGC) {
        int o = i / KGC, k = i % KGC;
        float s = 0.f;
        #pragma unroll
        for (int ss = 0; ss < SS; ++ss) s += wW[((size_t)(ss*CC + o))*KGC + k];
        wcomb[i] = s;
    } else if (i < CC*KGC + CC*KTC) {
        int j = i - CC*KGC;
        int o = j / KTC, k = j % KTC;
        int kt = k >> 6, c = k & 63;
        wre[j] = wt[((size_t)(o*CC + c))*9 + kt];
    }
}

// ---------------------------------------------------------------- fused GCN:
// gcn[n,o,t,v] = sum_{s,c,w} w_comb[o,s*64+c] * x[n,c,t,w] * adapt[n,s,w,v]
// Per block: one (n, 16-t tile). z slab (64 x 400) staged in LDS per s, then
// f32 WMMA GEMM  (M=64, N=400, K=192).  Biases dropped (cancel in BN1).
__global__ __launch_bounds__(1024)
void k_gcn(const float* __restrict__ x, const float* __restrict__ adapt,
           const float* __restrict__ wcomb, float* __restrict__ gcn)
{
    extern __shared__ float sm[];
    float* lds_w = sm;                    // 64*192  = 12288
    float* lds_z = sm + 12288;            // 64*400  = 25600
    float* lds_a = sm + 12288 + 25600;    // 3*25*25 =  1875
    const int n  = blockIdx.y;
    const int t0 = blockIdx.x * PT;
    const int tid = threadIdx.x;

    for (int i = tid; i < CC*KGC; i += 1024)   lds_w[i] = wcomb[i];
    for (int i = tid; i < SS*VV*VV; i += 1024) lds_a[i] = adapt[(size_t)n*SS*VV*VV + i];

    const int lane = tid & 31;
    const int wave = tid >> 5;
    const int wm   = wave >> 3;                     // o-tile 0..3
    const int wn   = wave & 7;                      // position-tile group 0..7
    const int tbase = (wn == 0) ? 0 : 4 + (wn-1)*3; // 25 n-tiles: 4,3,3,3,3,3,3,3
    const int tcnt  = (wn == 0) ? 4 : 3;
    const int lh  = lane >> 4;                      // lane half (K split)
    const int l15 = lane & 15;
    const int m   = wm*16 + l15;                    // A-matrix row (output channel)

    const int kc = tid >> 4;   // 0..63  = input channel c of this s-chunk
    const int tl = tid & 15;   // 0..15  = local time step

    v8f acc[4] = {};

    for (int s = 0; s < SS; ++s) {
        __syncthreads();
        // ---- VALU graph-mix: z[c][tl*25+v] = sum_w x[n,c,t0+tl,w]*adapt[s,w,v]
        {
            const int tau = t0 + tl;
            float xr[VV];
            if (tau < TDIM) {
                const float* xp = x + (((size_t)n*CC + kc)*TDIM + tau)*VV;
                #pragma unroll
                for (int w = 0; w < VV; ++w) xr[w] = xp[w];
            } else {
                #pragma unroll
                for (int w = 0; w < VV; ++w) xr[w] = 0.f;
            }
            float* zrow = lds_z + kc*PPOS + tl*VV;
            const float* ap = lds_a + s*VV*VV;
            for (int v = 0; v < VV; ++v) {
                float f = 0.f;
                #pragma unroll
                for (int w = 0; w < VV; ++w) f += xr[w] * ap[w*VV + v]; // LDS broadcast
                zrow[v] = f;
            }
        }
        __syncthreads();
        // ---- WMMA: acc += w_comb[:, s*64 + kl..] * z[kl..][:]
        for (int kl = 0; kl < 64; kl += 4) {
            // A-frag: two consecutive floats, 8B-aligned -> single ds_load_b64
            v2f a = *(const v2f*)(lds_w + m*KGC + s*64 + kl + 2*lh);
            #pragma unroll
            for (int j = 0; j < 4; ++j) {
                if (j < tcnt) {   // wave-uniform: EXEC all-1s or branch skipped
                    const int npos = (tbase + j)*16 + l15;
                    v2f b;
                    b.x = lds_z[(kl + 2*lh    )*PPOS + npos];
                    b.y = lds_z[(kl + 2*lh + 1)*PPOS + npos];
                    acc[j] = __builtin_amdgcn_wmma_f32_16x16x4_f32(
                        false, a, false, b, (short)0, acc[j], false, false);
                }
            }
        }
    }
    // ---- store D tiles (C/D layout: VGPR r -> M = r + 8*laneHalf, col = lane%16)
    #pragma unroll
    for (int j = 0; j < 4; ++j) {
        if (j < tcnt) {
            const int col = (tbase + j)*16 + l15;
            const int tau = t0 + col / VV;
            const int v   = col % VV;
            if (tau < TDIM) {
                #pragma unroll
                for (int r = 0; r < 8; ++r) {
                    int mo = wm*16 + r + lh*8;
                    gcn[(((size_t)n*CC + mo)*TDIM + tau)*VV + v] = acc[j][r];
                }
            }
        }
    }
}

// ---------------------------------------------------------------- BN stats:
// pass 1: per (channel, 4-batch slab) partial sum/sumsq (deterministic tree)
__global__ void k_stats_part(const float* __restrict__ in, double* __restrict__ part)
{
    __shared__ double s1[256], s2[256];
    const int slab = blockIdx.x;      // 0..15
    const int o    = blockIdx.y;      // channel
    const int tid  = threadIdx.x;
    double a = 0.0, b = 0.0;
    const int per = (NB/NSLAB)*TVS;   // 4*7500
    for (int i = tid; i < per; i += 256) {
        int nn = slab*(NB/NSLAB) + i / TVS;
        int r  = i % TVS;
        float v = in[((size_t)nn*CC + o)*TVS + r];
        a += v; b += (double)v * v;
    }
    s1[tid] = a; s2[tid] = b;
    __syncthreads();
    for (int st = 128; st > 0; st >>= 1) {
        if (tid < st) { s1[tid] += s1[tid+st]; s2[tid] += s2[tid+st]; }
        __syncthreads();
    }
    if (tid == 0) {
        part[((size_t)o*NSLAB + slab)*2    ] = s1[0];
        part[((size_t)o*NSLAB + slab)*2 + 1] = s2[0];
    }
}

// pass 2: fold slabs, produce fused per-channel scale/shift
__global__ void k_stats_fin(const double* __restrict__ part,
                            const float* __restrict__ gamma, const float* __restrict__ beta,
                            float* __restrict__ sc, float* __restrict__ sh)
{
    const int o = threadIdx.x;        // 64 threads
    double a = 0.0, b = 0.0;
    for (int s = 0; s < NSLAB; ++s) {
        a += part[((size_t)o*NSLAB + s)*2    ];
        b += part[((size_t)o*NSLAB + s)*2 + 1];
    }
    const double cnt = (double)NB * TVS;
    double mean = a / cnt;
    double var  = b / cnt - mean*mean;
    float scale = gamma[o] * (float)(1.0 / sqrt(var + 1e-5));
    sc[o] = scale;
    sh[o] = beta[o] - (float)mean * scale;
}

// ---------------------------------------------------------------- temporal conv:
// t[n,o,tau,v] = sum_{c,kt} w_t[o,c,kt] * relu(bn1(gcn))[n,c,tau+kt-4,v]
// Implicit GEMM, K = 576.  Input tile (with +/-4 halo) staged BN+ReLU'd in LDS
// (64x600 = 153.6 KB, fits CDNA5's 320KB WGP LDS).  b_t dropped (cancels in BN2).
__global__ __launch_bounds__(1024)
void k_tconv(const float* __restrict__ gcn, const float* __restrict__ wre,
             const float* __restrict__ sc1, const float* __restrict__ sh1,
             float* __restrict__ tout)
{
    extern __shared__ float gin[];  // [c][pp], pp in [0,600)
    const int n  = blockIdx.y;
    const int t0 = blockIdx.x * PT;
    const int tid = threadIdx.x;

    for (int i = tid; i < CC*600; i += 1024) {
        int c = i / 600, pp = i % 600;
        int tau = t0 - 4 + pp / VV;
        int v   = pp % VV;
        float val = 0.f;
        if (tau >= 0 && tau < TDIM) {
            float g = gcn[(((size_t)n*CC + c)*TDIM + tau)*VV + v];
            g = g * sc1[c] + sh1[c];
            val = g > 0.f ? g : 0.f;    // zero padding applied AFTER bn+relu (matches ref)
        }
        gin[i] = val;
    }
    __syncthreads();

    const int lane = tid & 31;
    const int wave = tid >> 5;
    const int wm   = wave >> 3;
    const int wn   = wave & 7;
    const int tbase = (wn == 0) ? 0 : 4 + (wn-1)*3;
    const int tcnt  = (wn == 0) ? 4 : 3;
    const int lh  = lane >> 4;
    const int l15 = lane & 15;
    const int m   = wm*16 + l15;

    v8f acc[4] = {};
    for (int k0 = 0; k0 < KTC; k0 += 4) {   // k = kt*64 + c ; 4-chunk never crosses kt
        const int kt = k0 >> 6;
        const int cb = k0 & 63;
        // A-frag: 8B-aligned pair from L2-resident reordered w_t -> global_load_b64
        v2f a = *(const v2f*)(wre + (size_t)m*KTC + k0 + 2*lh);
        #pragma unroll
        for (int j = 0; j < 4; ++j) {
            if (j < tcnt) {
                const int npos = (tbase + j)*16 + l15 + kt*VV; // p + kt*25 (halo shift)
                v2f b;
                b.x = gin[(cb + 2*lh    )*600 + npos];
                b.y = gin[(cb + 2*lh + 1)*600 + npos];
                acc[j] = __builtin_amdgcn_wmma_f32_16x16x4_f32(
                    false, a, false, b, (short)0, acc[j], false, false);
            }
        }
    }
    #pragma unroll
    for (int j = 0; j < 4; ++j) {
        if (j < tcnt) {
            const int col = (tbase + j)*16 + l15;
            const int tau = t0 + col / VV;
            const int v   = col % VV;
            if (tau < TDIM) {
                #pragma unroll
                for (int r = 0; r < 8; ++r) {
                    int mo = wm*16 + r + lh*8;
                    tout[(((size_t)n*CC + mo)*TDIM + tau)*VV + v] = acc[j][r];
                }
            }
        }
    }
}

// ---------------------------------------------------------------- final:
// out = relu(bn2(t) + x), float4-vectorized, in place on d_out
__global__ void k_final(const float* __restrict__ x, const float* __restrict__ sc2,
                        const float* __restrict__ sh2, float* __restrict__ out)
{
    size_t i = (size_t)blockIdx.x * blockDim.x + threadIdx.x;
    const size_t tot4 = (size_t)NB*CC*TVS/4;
    if (i >= tot4) return;
    size_t f = i*4;
    int o = (int)((f / TVS) % CC);   // 7500 % 4 == 0 -> channel uniform per float4
    float s = sc2[o], h = sh2[o];
    const float4* x4 = (const float4*)x;
    float4* o4 = (float4*)out;
    float4 t = o4[i], xv = x4[i], r;
    r.x = fmaxf(t.x*s + h + xv.x, 0.f);
    r.y = fmaxf(t.y*s + h + xv.y, 0.f);
    r.z = fmaxf(t.z*s + h + xv.z, 0.f);
    r.w = fmaxf(t.w*s + h + xv.w, 0.f);
    o4[i] = r;
}

// ----------------------------------------------------------------
extern "C" void kernel_launch(void* const* d_in, const int* in_sizes, int n_in,
                              void* d_out, int out_size, void* d_ws, size_t ws_size,
                              hipStream_t stream)
{
    (void)in_sizes; (void)n_in; (void)out_size; (void)ws_size;
    const float* x    = (const float*)d_in[0];
    const float* A    = (const float*)d_in[1];
    const float* wth  = (const float*)d_in[2];
    const float* bth  = (const float*)d_in[3];
    const float* wph  = (const float*)d_in[4];
    const float* bph  = (const float*)d_in[5];
    const float* Bm   = (const float*)d_in[6];
    const float* wW   = (const float*)d_in[7];
    /* d_in[8] = b_W : per-channel constant, exactly cancelled by BN1 mean-sub */
    const float* gs   = (const float*)d_in[9];
    const float* bs   = (const float*)d_in[10];
    const float* wt   = (const float*)d_in[11];
    /* d_in[12] = b_t : per-channel constant, exactly cancelled by BN2 mean-sub */
    const float* gt   = (const float*)d_in[13];
    const float* bt_  = (const float*)d_in[14];

    float* ws    = (float*)d_ws;
    float* theta = ws;                  // 102400
    float* phi   = theta + 102400;      // 102400
    float* adapt = phi   + 102400;      // 120000
    float* wcomb = adapt + 120000;      // 12288
    float* wre   = wcomb + 12288;       // 36864
    float* sc1   = wre   + 36864;       // 64
    float* sh1   = sc1 + 64;
    float* sc2   = sh1 + 64;
    float* sh2   = sc2 + 64;            // offset 374208 floats (8B aligned)
    double* part = (double*)(sh2 + 64); // 64*16*2 doubles = 4096 floats
    float* gcn   = sh2 + 64 + 4096;     // 30,720,000  (total ~124.4 MB)

    float* tbuf  = (float*)d_out;       // temporal-conv result lives in d_out

    k_stage1  <<<NB, 256, 0, stream>>>(x, wth, bth, wph, bph, theta, phi);
    k_adaptive<<<NB, 256, 0, stream>>>(theta, phi, A, Bm, adapt);
    k_prep    <<<(CC*KGC + CC*KTC + 255)/256, 256, 0, stream>>>(wW, wt, wcomb, wre);

    const size_t gcn_lds = (size_t)(12288 + 25600 + 1875) * sizeof(float); // 159 KB
    k_gcn<<<dim3(NTIL, NB), 1024, gcn_lds, stream>>>(x, adapt, wcomb, gcn);

    k_stats_part<<<dim3(NSLAB, CC), 256, 0, stream>>>(gcn, part);
    k_stats_fin <<<1, CC, 0, stream>>>(part, gs, bs, sc1, sh1);

    const size_t tc_lds = (size_t)(CC*600) * sizeof(float);                // 153.6 KB
    k_tconv<<<dim3(NTIL, NB), 1024, tc_lds, stream>>>(gcn, wre, sc1, sh1, tbuf);

    k_stats_part<<<dim3(NSLAB, CC), 256, 0, stream>>>(tbuf, part);
    k_stats_fin <<<1, CC, 0, stream>>>(part, gt, bt_, sc2, sh2);

    const size_t tot4 = (size_t)NB*CC*TVS/4;
    k_final<<<(unsigned)((tot4 + 255)/256), 256, 0, stream>>>(x, sc2, sh2, (float*)d_out);
}